// FilterLayer_5360119186018
// MI455X (gfx1250) — compile-verified
//
#include <hip/hip_runtime.h>
#include <hip/hip_bf16.h>
#include <stdint.h>

// Problem constants (from reference): N=16, L=4096, D=1024, k = ceil(L*0.5) = 2048
#define LEN   4096
#define KSEL  2048
#define DIM   1024
#define NEGV  (-1000.0f)

// ---------------------------------------------------------------------------
// Kernel 1: per-row top-k via LDS bitonic sort of 64-bit keys.
// key = (monotonic_u32(score) << 32) | (0xFFFFFFFF - index)
// Descending sort => highest score first; ties => lowest index first
// (matches jax.lax.top_k stability).
// ---------------------------------------------------------------------------
__global__ __launch_bounds__(1024) void topk_select_kernel(
    const float* __restrict__ cls_attn,          // [N, L]
    const unsigned char* __restrict__ attn_mask, // [N, L] (jnp.bool_ = 1 byte)
    const int* __restrict__ cls_indices,         // [N]
    int* __restrict__ out_idx)                   // [N, KSEL]
{
    __shared__ unsigned long long keys[LEN];     // 32 KB of 320 KB LDS

    const int n   = blockIdx.x;
    const int tid = threadIdx.x;                 // 1024 threads
    const int ci  = cls_indices[n];

    const float* row_attn = cls_attn + (size_t)n * LEN;
    const unsigned char* row_mask = attn_mask + (size_t)n * LEN;

    // Build keys
    for (int i = tid; i < LEN; i += 1024) {
        float s = row_attn[i];
        if (row_mask[i] != 0 || i == ci) s = NEGV;
        unsigned u = __float_as_uint(s);
        u = (u & 0x80000000u) ? ~u : (u | 0x80000000u);   // monotonic map
        keys[i] = ((unsigned long long)u << 32) | (unsigned)(0xFFFFFFFFu - (unsigned)i);
    }
    __syncthreads();

    // Bitonic sort, descending. 2048 compare pairs per step; 1024 threads x 2.
    for (unsigned k = 2; k <= LEN; k <<= 1) {
        for (unsigned j = k >> 1; j >= 1; j >>= 1) {
            for (unsigned t = tid; t < (LEN / 2); t += 1024) {
                unsigned i = ((t & ~(j - 1u)) << 1) | (t & (j - 1u));
                unsigned p = i | j;
                bool desc = ((i & k) == 0);
                unsigned long long a = keys[i];
                unsigned long long b = keys[p];
                if (desc ? (a < b) : (a > b)) {
                    keys[i] = b;
                    keys[p] = a;
                }
            }
            __syncthreads();
        }
    }

    // Emit top-KSEL indices (descending-score order)
    for (int jj = tid; jj < KSEL; jj += 1024) {
        unsigned lo = (unsigned)(keys[jj] & 0xFFFFFFFFu);
        out_idx[(size_t)n * KSEL + jj] = (int)(0xFFFFFFFFu - lo);
    }
}

// ---------------------------------------------------------------------------
// Kernel 2: gather selected token rows (4 KB each) via CDNA5 async
// global<->LDS DMA. One wave32 per token row: 8 x b128 async loads cover
// 4 KB (32 lanes x 16 B x 8), wait ASYNCcnt, 8 x b128 async stores.
// Per the ISA pseudocode the instruction offset advances BOTH the global
// and LDS byte addresses, so one base pair serves all 8 chunks.
// Streaming traffic with zero reuse -> non-temporal hints keep L2 clean.
// Final ASYNCcnt drain is implicit in S_ENDPGM's S_WAIT_IDLE.
// ---------------------------------------------------------------------------
#define ASYNC_LD(off) \
    asm volatile("global_load_async_to_lds_b128 %0, %1, off offset:" #off \
                 " th:TH_LOAD_NT" \
                 :: "v"(lds_addr), "v"(gsrc) : "memory")
#define ASYNC_ST(off) \
    asm volatile("global_store_async_from_lds_b128 %0, %1, off offset:" #off \
                 " th:TH_STORE_NT" \
                 :: "v"(gdst), "v"(lds_addr) : "memory")

__global__ __launch_bounds__(256) void gather_async_kernel(
    const float* __restrict__ x,        // [N, L, D]
    const int* __restrict__ idxbuf,     // [N, KSEL]
    float* __restrict__ out)            // [N, KSEL, D]
{
    __shared__ __align__(16) unsigned char stage[8 * 4096]; // 4 KB per wave

    const int wave = threadIdx.x >> 5;          // 0..7
    const int lane = threadIdx.x & 31;
    const int r    = blockIdx.x * 8 + wave;     // global row id, 0..N*KSEL-1
    const int n    = r >> 11;                   // r / KSEL
    const int idx  = idxbuf[r];

    const float* src = x + ((size_t)n * LEN + (size_t)idx) * DIM;
    float*       dst = out + (size_t)r * DIM;

    // LDS byte offset of this wave's staging buffer + this lane's 16B slot.
    // Low 32 bits of a generic pointer into LDS are the LDS byte address.
    unsigned lds_addr =
        (unsigned)(uintptr_t)(&stage[wave * 4096]) + (unsigned)(lane * 16);

    unsigned long long gsrc = (unsigned long long)(uintptr_t)src + (unsigned)(lane * 16);
    unsigned long long gdst = (unsigned long long)(uintptr_t)dst + (unsigned)(lane * 16);

    // 8 chunks x 512 B = 4 KB row, memory -> LDS
    ASYNC_LD(0);    ASYNC_LD(512);  ASYNC_LD(1024); ASYNC_LD(1536);
    ASYNC_LD(2048); ASYNC_LD(2560); ASYNC_LD(3072); ASYNC_LD(3584);
    asm volatile("s_wait_asynccnt 0" ::: "memory");   // LDS data now visible

    // LDS -> memory (drained implicitly by S_ENDPGM's wait-idle)
    ASYNC_ST(0);    ASYNC_ST(512);  ASYNC_ST(1024); ASYNC_ST(1536);
    ASYNC_ST(2048); ASYNC_ST(2560); ASYNC_ST(3072); ASYNC_ST(3584);
}

// ---------------------------------------------------------------------------
// Launch
// ---------------------------------------------------------------------------
extern "C" void kernel_launch(void* const* d_in, const int* in_sizes, int n_in,
                              void* d_out, int out_size, void* d_ws, size_t ws_size,
                              hipStream_t stream) {
    const float*         x           = (const float*)d_in[0];
    const float*         cls_attn    = (const float*)d_in[1];
    const unsigned char* attn_mask   = (const unsigned char*)d_in[2];
    const int*           cls_indices = (const int*)d_in[3];
    float*               out         = (float*)d_out;

    const int N = in_sizes[3];            // 16

    int* topk_idx = (int*)d_ws;           // N*KSEL ints = 128 KB scratch

    // 1) select top-k indices per row
    topk_select_kernel<<<N, 1024, 0, stream>>>(cls_attn, attn_mask, cls_indices,
                                               topk_idx);

    // 2) gather: one wave per token row; 8 waves per block
    const int total_rows = N * KSEL;          // 32768
    const int blocks     = total_rows / 8;    // 4096
    gather_async_kernel<<<blocks, 256, 0, stream>>>(x, topk_idx, out);
}